// SpeAttention_6227702579309
// MI455X (gfx1250) — compile-verified
//
#include <hip/hip_runtime.h>
#include <hip/hip_bf16.h>
#include <float.h>
#include <math.h>

// ---- problem constants (from reference) ----
#define BATCH   16
#define LSEQ    128          // sequence length == Cin
#define DM      32           // D_MODEL
#define NS      8            // D_STATE
#define HWPIX   (256*256)    // H*W per (b,c) row
#define SPLIT   4            // blocks per reduction row

typedef __attribute__((ext_vector_type(2))) float v2f;
typedef __attribute__((ext_vector_type(4))) float v4f;
typedef __attribute__((ext_vector_type(8))) float v8f;

// =====================================================================
// Kernel 1: s[b,c] = max over H,W  (bandwidth-bound: 536 MB @ 23.3 TB/s)
// =====================================================================
__global__ __launch_bounds__(256)
void max_reduce_kernel(const float* __restrict__ x, float* __restrict__ partials) {
    const int blk = blockIdx.x;            // BATCH*LSEQ*SPLIT blocks
    const int row = blk / SPLIT;
    const int seg = blk % SPLIT;
    const int tid = threadIdx.x;
    const int seglen4 = (HWPIX / SPLIT) / 4;   // 4096 float4 per segment
    const v4f* p = (const v4f*)(x + (size_t)row * HWPIX + (size_t)seg * (HWPIX / SPLIT));
    float m = -FLT_MAX;
    for (int i = tid; i < seglen4; i += 256) {
        v4f v = __builtin_nontemporal_load(&p[i]);   // stream, don't pollute L2
        m = fmaxf(m, fmaxf(fmaxf(v[0], v[1]), fmaxf(v[2], v[3])));
    }
    __shared__ float red[256];
    red[tid] = m;
    __syncthreads();
    for (int s = 128; s > 0; s >>= 1) {
        if (tid < s) red[tid] = fmaxf(red[tid], red[tid + s]);
        __syncthreads();
    }
    if (tid == 0) partials[row * SPLIT + seg] = red[0];
}

// =====================================================================
// helpers
// =====================================================================
__device__ __forceinline__ float siluf(float v)     { return v / (1.0f + expf(-v)); }
__device__ __forceinline__ float softplusf(float v) { return (v > 20.0f) ? v : log1pf(expf(v)); }

// 16x16 fp32 WMMA tile accumulate over K=32:
// C[M0+v+8*hi][N via lane] += sum_k Arow(M=lane&15,k) * Brow(N=lane&15,k)
// where caller supplies per-lane row pointers for A (stride sA) and B (stride sB).
__device__ __forceinline__ v8f wmma_k32_f32(const float* __restrict__ Arow, int sA,
                                            const float* __restrict__ Brow, int sB,
                                            int lane) {
    v8f acc = {};
    const int hi = lane >> 4;          // 0 or 1
#if defined(__has_builtin) && __has_builtin(__builtin_amdgcn_wmma_f32_16x16x4_f32)
    #pragma unroll
    for (int kk = 0; kk < 8; ++kk) {
        const int kb = kk * 4 + hi * 2;          // ISA layout: K = vgpr + 2*(lane>=16)
        v2f a, b;
        a[0] = Arow[kb];  a[1] = Arow[kb + 1];
        b[0] = Brow[kb];  b[1] = Brow[kb + 1];
        acc = __builtin_amdgcn_wmma_f32_16x16x4_f32(false, a, false, b,
                                                    (short)0, acc, false, false);
    }
    (void)sA; (void)sB;
#else
    // scalar fallback producing identical lane-local results (C/D layout emulated)
    const int lr = lane & 15;
    #pragma unroll
    for (int v = 0; v < 8; ++v) {
        float s = 0.0f;
        const float* Ar = Arow - (long)(lane & 15) * sA + (long)(v + 8 * hi) * sA; (void)lr;
        for (int k = 0; k < 32; ++k) s += Ar[k] * Brow[k];
        acc[v] = s;
    }
#endif
    return acc;
}

// =====================================================================
// Kernel 2: full per-batch pipeline in LDS. One block per batch.
//   LDS pool (floats):
//     XZ  : 64 x 128                @ 0       (8192)  [also YLN/Y2 later]
//     U1  : 32 x 128 (u buffer)     @ 8192    (4096)
//     U2  : h[l][33] then OUT[d][l] @ 12288   (4224)
//     XD  : 128 x 20 (dt|B|C)       @ 16512   (2560)
//     SL  : 128                     @ 19072   (128)
//   total 19200 floats = 76800 B dynamic LDS (gfx1250: up to 320 KB/WG)
// =====================================================================
#define OFF_XZ  0
#define OFF_U1  8192
#define OFF_U2  12288
#define OFF_XD  16512
#define OFF_SL  19072
#define POOL_FLOATS 19200
#define OFF_YLN 0        // aliases dead XZ
#define OFF_Y2  4224     // aliases dead XZ (+ spills 256 floats into dead U1)

__device__ void ssm_branch(int tid, bool flip, bool accumulate,
                           float* __restrict__ XZ, float* __restrict__ bufU,
                           float* __restrict__ OUT, float* __restrict__ XD,
                           const float* __restrict__ convw, const float* __restrict__ convb,
                           const float* __restrict__ xproj, const float* __restrict__ dtw,
                           const float* __restrict__ dtb,   const float* __restrict__ Alog,
                           const float* __restrict__ Dvec) {
    // --- u = silu(causal depthwise conv) ---
    for (int idx = tid; idx < DM * LSEQ; idx += 256) {
        const int d = idx >> 7;          // idx / 128
        const int l = idx & 127;
        float acc = convb[d];
        #pragma unroll
        for (int j = 0; j < 4; ++j) {
            const int li = l - 3 + j;
            float xv = 0.0f;
            if (li >= 0) xv = XZ[d * LSEQ + (flip ? (LSEQ - 1 - li) : li)];
            acc += convw[d * 4 + j] * xv;
        }
        bufU[d * LSEQ + l] = siluf(acc);
    }
    __syncthreads();
    // --- x_dbl[l][r] = sum_d u[d][l] * xproj[r][d]  (r < 18) ---
    for (int idx = tid; idx < LSEQ * 18; idx += 256) {
        const int l = idx / 18, r = idx % 18;
        float acc = 0.0f;
        for (int d = 0; d < DM; ++d) acc += bufU[d * LSEQ + l] * xproj[r * DM + d];
        XD[l * 20 + r] = acc;
    }
    __syncthreads();
    // --- selective scan: thread -> (d, n) state element ---
    {
        const int d = tid >> 3;          // 0..31
        const int n = tid & 7;           // 0..7
        const float A_dn = -expf(Alog[d * NS + n]);
        const float Wdt0 = dtw[d * 2 + 0], Wdt1 = dtw[d * 2 + 1];
        const float bdt  = dtb[d];
        const float Dp   = Dvec[d];
        float hstate = 0.0f;
        for (int l = 0; l < LSEQ; ++l) {
            const float dt0 = XD[l * 20 + 0];
            const float dt1 = XD[l * 20 + 1];
            const float Bn  = XD[l * 20 + 2 + n];
            const float Cn  = XD[l * 20 + 10 + n];
            const float delta = softplusf(dt0 * Wdt0 + dt1 * Wdt1 + bdt);
            const float u_dl  = bufU[d * LSEQ + l];
            hstate = expf(delta * A_dn) * hstate + delta * u_dl * Bn;
            float part = hstate * Cn;
            part += __shfl_xor(part, 1, 32);     // reduce over n (8 lanes, in-wave)
            part += __shfl_xor(part, 2, 32);
            part += __shfl_xor(part, 4, 32);
            if (n == 0) {
                const float y  = part + Dp * u_dl;
                const float zv = XZ[(DM + d) * LSEQ + (flip ? (LSEQ - 1 - l) : l)];
                const float val = y * siluf(zv);
                if (accumulate) OUT[d * LSEQ + (LSEQ - 1 - l)] += val;  // un-flip on the fly
                else            OUT[d * LSEQ + l] = val;
            }
        }
    }
    __syncthreads();
}

__global__ __launch_bounds__(256)
void mamba_pipeline_kernel(const float* __restrict__ partials,
                           const float* __restrict__ w1,     const float* __restrict__ b1,
                           const float* __restrict__ ln_g,   const float* __restrict__ ln_b,
                           const float* __restrict__ in_proj,
                           const float* __restrict__ convw_f, const float* __restrict__ convb_f,
                           const float* __restrict__ xproj_f, const float* __restrict__ dtw_f,
                           const float* __restrict__ dtb_f,   const float* __restrict__ Alog_f,
                           const float* __restrict__ D_f,
                           const float* __restrict__ convw_b, const float* __restrict__ convb_b,
                           const float* __restrict__ xproj_b, const float* __restrict__ dtw_b,
                           const float* __restrict__ dtb_b,   const float* __restrict__ Alog_b,
                           const float* __restrict__ D_b,
                           const float* __restrict__ norm_g,  const float* __restrict__ norm_b,
                           const float* __restrict__ outw,    const float* __restrict__ w2,
                           const float* __restrict__ b2,
                           float* __restrict__ out) {
    extern __shared__ float pool[];
    float* XZ  = pool + OFF_XZ;
    float* U1  = pool + OFF_U1;
    float* U2  = pool + OFF_U2;
    float* XD  = pool + OFF_XD;
    float* SL  = pool + OFF_SL;
    float* YLN = pool + OFF_YLN;
    float* Y2  = pool + OFF_Y2;

    const int tid  = threadIdx.x;
    const int b    = blockIdx.x;
    const int wv   = tid >> 5;       // wave id 0..7
    const int lane = tid & 31;
    const int lr   = lane & 15;
    const int hi   = lane >> 4;

    // ---- phase A: combine reduction partials, h = s*w1+b1, layernorm over d ----
    if (tid < LSEQ) {
        const float* pp = partials + (b * LSEQ + tid) * SPLIT;
        SL[tid] = fmaxf(fmaxf(pp[0], pp[1]), fmaxf(pp[2], pp[3]));
    }
    __syncthreads();
    if (tid < LSEQ) {
        const int l = tid;
        const float s = SL[l];
        float tmp[DM];
        float m = 0.0f;
        for (int d = 0; d < DM; ++d) { tmp[d] = s * w1[d] + b1[d]; m += tmp[d]; }
        m *= (1.0f / DM);
        float vv = 0.0f;
        for (int d = 0; d < DM; ++d) { const float df = tmp[d] - m; vv += df * df; }
        vv *= (1.0f / DM);
        const float rs = rsqrtf(vv + 1e-5f);
        for (int d = 0; d < DM; ++d)
            U2[l * 33 + d] = (tmp[d] - m) * rs * ln_g[d] + ln_b[d];   // h[l][33]
    }
    __syncthreads();

    // ---- phase B: xz[e][l] = sum_d in_proj[e][d] * h[l][d]  via V_WMMA_F32_16X16X4_F32 ----
    // 32 tiles of 16x16, K=32 -> 8 WMMA steps each; 4 tiles per wave.
    for (int t = wv; t < 32; t += 8) {
        const int E0 = (t >> 3) << 4;
        const int L0 = (t & 7) << 4;
        v8f acc = wmma_k32_f32(in_proj + (E0 + lr) * DM, DM,     // A: M=e rows
                               U2      + (L0 + lr) * 33, 33,     // B: B[k][n]=h[n][k]
                               lane);
        #pragma unroll
        for (int v = 0; v < 8; ++v)
            XZ[(E0 + v + 8 * hi) * LSEQ + L0 + lr] = acc[v];
    }
    __syncthreads();

    // ---- phase C: forward + backward SSM branches (backward accumulates, un-flipped) ----
    ssm_branch(tid, false, false, XZ, U1, U2, XD,
               convw_f, convb_f, xproj_f, dtw_f, dtb_f, Alog_f, D_f);
    ssm_branch(tid, true, true, XZ, U1, U2, XD,
               convw_b, convb_b, xproj_b, dtw_b, dtb_b, Alog_b, D_b);

    // ---- phase D: layernorm over d, * 0.5 ----
    if (tid < LSEQ) {
        const int l = tid;
        float tmp[DM];
        float m = 0.0f;
        for (int d = 0; d < DM; ++d) { tmp[d] = U2[d * LSEQ + l]; m += tmp[d]; }
        m *= (1.0f / DM);
        float vv = 0.0f;
        for (int d = 0; d < DM; ++d) { const float df = tmp[d] - m; vv += df * df; }
        vv *= (1.0f / DM);
        const float rs = rsqrtf(vv + 1e-5f);
        for (int d = 0; d < DM; ++d)
            YLN[l * 33 + d] = ((tmp[d] - m) * rs * norm_g[d] + norm_b[d]) * 0.5f;
    }
    __syncthreads();

    // ---- phase E: y2[l][o] = sum_d yln[l][d] * outw[o][d]  via WMMA (16 tiles, 2/wave) ----
    for (int t = wv; t < 16; t += 8) {
        const int L0 = (t >> 1) << 4;
        const int O0 = (t & 1) << 4;
        v8f acc = wmma_k32_f32(YLN  + (L0 + lr) * 33, 33,        // A: M=l rows
                               outw + (O0 + lr) * DM, DM,        // B: B[k][n]=outw[n][k]
                               lane);
        #pragma unroll
        for (int v = 0; v < 8; ++v)
            Y2[(L0 + v + 8 * hi) * 33 + O0 + lr] = acc[v];
    }
    __syncthreads();

    // ---- phase F: o[l] = sigmoid(b2 + sum_o y2[l][o] * w2[o]) ----
    if (tid < LSEQ) {
        const int l = tid;
        float acc = b2[0];
        for (int o = 0; o < DM; ++o) acc += Y2[l * 33 + o] * w2[o];
        out[b * LSEQ + l] = 1.0f / (1.0f + expf(-acc));
    }
}

// =====================================================================
// launch
// =====================================================================
extern "C" void kernel_launch(void* const* d_in, const int* in_sizes, int n_in,
                              void* d_out, int out_size, void* d_ws, size_t ws_size,
                              hipStream_t stream) {
    const float* x       = (const float*)d_in[0];
    const float* w1      = (const float*)d_in[1];
    const float* b1      = (const float*)d_in[2];
    const float* ln_g    = (const float*)d_in[3];
    const float* ln_b    = (const float*)d_in[4];
    const float* in_proj = (const float*)d_in[5];
    const float* convw_f = (const float*)d_in[6];
    const float* convb_f = (const float*)d_in[7];
    const float* xproj_f = (const float*)d_in[8];
    const float* dtw_f   = (const float*)d_in[9];
    const float* dtb_f   = (const float*)d_in[10];
    const float* Alog_f  = (const float*)d_in[11];
    const float* D_f     = (const float*)d_in[12];
    const float* convw_b = (const float*)d_in[13];
    const float* convb_b = (const float*)d_in[14];
    const float* xproj_b = (const float*)d_in[15];
    const float* dtw_b   = (const float*)d_in[16];
    const float* dtb_b   = (const float*)d_in[17];
    const float* Alog_b  = (const float*)d_in[18];
    const float* D_b     = (const float*)d_in[19];
    const float* norm_g  = (const float*)d_in[20];
    const float* norm_b  = (const float*)d_in[21];
    const float* outw    = (const float*)d_in[22];
    const float* w2      = (const float*)d_in[23];
    const float* b2      = (const float*)d_in[24];
    float* out      = (float*)d_out;
    float* partials = (float*)d_ws;     // BATCH*LSEQ*SPLIT floats = 32 KB

    // Kernel 1: the bandwidth-bound spatial max (dominates runtime).
    max_reduce_kernel<<<BATCH * LSEQ * SPLIT, 256, 0, stream>>>(x, partials);

    // Kernel 2: whole per-batch Mamba pipeline in LDS, WMMA for the GEMMs.
    mamba_pipeline_kernel<<<BATCH, 256, POOL_FLOATS * sizeof(float), stream>>>(
        partials, w1, b1, ln_g, ln_b, in_proj,
        convw_f, convb_f, xproj_f, dtw_f, dtb_f, Alog_f, D_f,
        convw_b, convb_b, xproj_b, dtw_b, dtb_b, Alog_b, D_b,
        norm_g, norm_b, outw, w2, b2, out);
}